// MPNN_40484361732177
// MI455X (gfx1250) — compile-verified
//
#include <hip/hip_runtime.h>
#include <hip/hip_bf16.h>
#include <limits.h>

typedef __attribute__((ext_vector_type(2))) float v2f;
typedef __attribute__((ext_vector_type(8))) float v8f;

#define DDIM 128
#define HHEADS 4
#define HF 512      // H * F
#define NEG_SLOPE 0.2f

// ---------- helpers ----------
__device__ __forceinline__ int enc_f(float x) {
    int i = __float_as_int(x);
    return (i >= 0) ? i : (i ^ 0x7FFFFFFF);
}
__device__ __forceinline__ float dec_f(int i) {
    return __int_as_float((i >= 0) ? i : (i ^ 0x7FFFFFFF));
}

__global__ void fill_int_kernel(int* p, int v, int n) {
    int i = blockIdx.x * blockDim.x + threadIdx.x;
    if (i < n) p[i] = v;
}

// ---------- WMMA GEMM:  C[M,Kout] = A[M,128] @ B[128,Kout] (+ bias[Kout]) ----------
// One 16x16 output tile per wave; 32 x v_wmma_f32_16x16x4_f32 (exact fp32).
// OOB M-rows: A address is CLAMPED (row m of A only feeds row m of D, never
// stored when m >= M) -> branch-free inner loop. Epilogue takes a wave-uniform
// fast path (unconditional stores) for full tiles; only the single ragged
// M-tile uses per-lane guarded stores.
__global__ void wmma_gemm_kernel(const float* __restrict__ A,
                                 const float* __restrict__ B,
                                 const float* __restrict__ bias,
                                 float* __restrict__ C,
                                 int M, int Kout) {
    const int tilesN = Kout >> 4;
    const int tilesM = (M + 15) >> 4;
    int wave = blockIdx.x * (blockDim.x >> 5) + (threadIdx.x >> 5);
    if (wave >= tilesM * tilesN) return;          // wave-uniform: EXEC all-1s below
    int tM = wave / tilesN;
    int tN = wave - tM * tilesN;

    int lane = threadIdx.x & 31;
    int half = lane >> 4;          // 0: K = k0,k0+1   1: K = k0+2,k0+3
    int lm   = lane & 15;

    int rowA  = tM * 16 + lm;                     // A-matrix M index for this lane
    int rowAc = (rowA < M) ? rowA : (M - 1);      // clamp: memory-safe, result discarded
    const float* Arow = A + (size_t)rowAc * DDIM + 2 * half;
    int colB = tN * 16 + lm;                      // B/C N index for this lane
    const float* Bcol = B + (size_t)(2 * half) * Kout + colB;

    v8f acc = {0.f, 0.f, 0.f, 0.f, 0.f, 0.f, 0.f, 0.f};

#pragma unroll 4
    for (int k0 = 0; k0 < DDIM; k0 += 4) {
        v2f a;
        a.x = Arow[k0];                           // contiguous pair -> b64 load
        a.y = Arow[k0 + 1];
        v2f b;
        b.x = Bcol[(size_t)k0 * Kout];
        b.y = Bcol[(size_t)(k0 + 1) * Kout];
        acc = __builtin_amdgcn_wmma_f32_16x16x4_f32(
            /*neg_a=*/false, a, /*neg_b=*/false, b,
            /*c_mod=*/(short)0, acc, /*reuse_a=*/false, /*reuse_b=*/false);
    }

    float bv = bias ? bias[colB] : 0.0f;
    float* Cl = C + (size_t)(tM * 16 + 8 * half) * Kout + colB;
    if (tM * 16 + 16 <= M) {
        // full tile: unconditional, branch-free stores
#pragma unroll
        for (int v = 0; v < 8; ++v)
            Cl[(size_t)v * Kout] = acc[v] + bv;
    } else {
        // ragged last M-tile only
#pragma unroll
        for (int v = 0; v < 8; ++v) {
            int rm = tM * 16 + v + 8 * half;
            if (rm < M) Cl[(size_t)v * Kout] = acc[v] + bv;
        }
    }
}

// ---------- degrees ----------
__global__ void deg_kernel(const int* __restrict__ row, const int* __restrict__ col,
                           const float* __restrict__ av,
                           float* edge_deg, float* cntB, float* DnRaw, int nnz) {
    int i = blockIdx.x * blockDim.x + threadIdx.x;
    if (i >= nnz) return;
    int c = col[i], r = row[i];
    atomicAdd(&edge_deg[c], av[i]);
    atomicAdd(&cntB[c], 1.0f);
    atomicAdd(&DnRaw[r], av[c]);   // reference: adj_values[col] scattered by row
}

__global__ void inv_kernel(float* p, int n) {   // p = (p>0) ? 1/p : 0
    int i = blockIdx.x * blockDim.x + threadIdx.x;
    if (i < n) { float v = p[i]; p[i] = (v > 0.f) ? 1.0f / v : 0.0f; }
}

// ---------- agg[e,:] += av[i] * X_trans[row[i],:]  (wave per nnz) ----------
__global__ void scatter_agg_kernel(const int* __restrict__ row, const int* __restrict__ col,
                                   const float* __restrict__ av,
                                   const float* __restrict__ Xt, float* agg, int nnz) {
    int w = blockIdx.x * (blockDim.x >> 5) + (threadIdx.x >> 5);
    if (w >= nnz) return;
    int lane = threadIdx.x & 31;
    int r = row[w], c = col[w];
    float v = av[w];
    float4 xv = *(const float4*)(Xt + (size_t)r * DDIM + lane * 4);
    float* dst = agg + (size_t)c * DDIM + lane * 4;
    atomicAdd(dst + 0, v * xv.x);
    atomicAdd(dst + 1, v * xv.y);
    atomicAdd(dst + 2, v * xv.z);
    atomicAdd(dst + 3, v * xv.w);
}

// ---------- E_res = agg/deg + Etmp + E ----------
__global__ void eres_kernel(const float* __restrict__ agg, const float* __restrict__ edge_deg,
                            const float* __restrict__ Etmp, const float* __restrict__ E,
                            float* __restrict__ Eres, int en) {
    int i = blockIdx.x * blockDim.x + threadIdx.x;
    if (i < en * DDIM) {
        int e = i >> 7;
        Eres[i] = agg[i] / edge_deg[e] + Etmp[i] + E[i];
    }
}

// ---------- out[m,h] = feat[m,h,:] . att[h, off:off+128] ----------
__global__ void attdot_kernel(const float* __restrict__ feat, const float* __restrict__ att,
                              int offF, float* __restrict__ out, int M) {
    int t = blockIdx.x * blockDim.x + threadIdx.x;
    if (t >= M * HHEADS) return;
    int h = t & 3, n = t >> 2;
    const float* fp = feat + (size_t)n * HF + h * DDIM;
    const float* ap = att + h * (2 * DDIM) + offF;
    float s = 0.f;
#pragma unroll 4
    for (int f = 0; f < DDIM; ++f) s += fp[f] * ap[f];
    out[t] = s;
}

// ---------- softmax pass 1: raw alpha + segment max ----------
__global__ void alpha_max_kernel(const int* __restrict__ row, const int* __restrict__ col,
                                 const float* __restrict__ xa, const float* __restrict__ ea,
                                 float* alpha, int* menc, int nnz) {
    int t = blockIdx.x * blockDim.x + threadIdx.x;
    if (t >= nnz * HHEADS) return;
    int h = t & 3, i = t >> 2;
    float a = xa[row[i] * HHEADS + h] + ea[col[i] * HHEADS + h];
    a = (a > 0.f) ? a : NEG_SLOPE * a;   // leaky relu
    alpha[t] = a;
    atomicMax(&menc[row[i] * HHEADS + h], enc_f(a));
}

// ---------- softmax pass 2: exp + segment sum ----------
__global__ void alpha_exp_kernel(const int* __restrict__ row, float* alpha,
                                 const int* __restrict__ menc, float* ssum, int nnz) {
    int t = blockIdx.x * blockDim.x + threadIdx.x;
    if (t >= nnz * HHEADS) return;
    int h = t & 3, i = t >> 2;
    float m = dec_f(menc[row[i] * HHEADS + h]);
    float ex = __expf(alpha[t] - m);
    alpha[t] = ex;
    atomicAdd(&ssum[row[i] * HHEADS + h], ex);
}

// ---------- softmax pass 3: normalize ----------
__global__ void alpha_norm_kernel(const int* __restrict__ row, float* alpha,
                                  const float* __restrict__ ssum, int nnz) {
    int t = blockIdx.x * blockDim.x + threadIdx.x;
    if (t >= nnz * HHEADS) return;
    int h = t & 3, i = t >> 2;
    alpha[t] /= ssum[row[i] * HHEADS + h];
}

// ---------- m1: out_e[col,h,:] += alpha[i,h] * x[row,h,:]   (Bn deferred) ----------
__global__ void m1_kernel(const int* __restrict__ row, const int* __restrict__ col,
                          const float* __restrict__ alpha, const float* __restrict__ x,
                          float* out_e, int nnz) {
    int w = blockIdx.x * (blockDim.x >> 5) + (threadIdx.x >> 5);
    if (w >= nnz) return;
    int lane = threadIdx.x & 31;
    int r = row[w], c = col[w];
    const float* xr = x + (size_t)r * HF;
    float* oe = out_e + (size_t)c * HF;
#pragma unroll
    for (int h = 0; h < HHEADS; ++h) {
        float al = alpha[w * HHEADS + h];
        float4 xv = *(const float4*)(xr + h * DDIM + lane * 4);
        float* d = oe + h * DDIM + lane * 4;
        atomicAdd(d + 0, al * xv.x);
        atomicAdd(d + 1, al * xv.y);
        atomicAdd(d + 2, al * xv.z);
        atomicAdd(d + 3, al * xv.w);
    }
}

// ---------- m2: X_res[row,:] += Dn[row]*Bn[col]*(1/H) * sum_h alpha[i,h]*out_e[col,h,:] ----------
__global__ void m2_kernel(const int* __restrict__ row, const int* __restrict__ col,
                          const float* __restrict__ alpha, const float* __restrict__ out_e,
                          const float* __restrict__ DnInv, const float* __restrict__ BnInv,
                          float* Xres, int nnz) {
    int w = blockIdx.x * (blockDim.x >> 5) + (threadIdx.x >> 5);
    if (w >= nnz) return;
    int lane = threadIdx.x & 31;
    int r = row[w], c = col[w];
    float scale = DnInv[r] * BnInv[c] * (1.0f / HHEADS);
    const float* oe = out_e + (size_t)c * HF + lane * 4;
    float4 acc = make_float4(0.f, 0.f, 0.f, 0.f);
#pragma unroll
    for (int h = 0; h < HHEADS; ++h) {
        float al = alpha[w * HHEADS + h];
        float4 ov = *(const float4*)(oe + h * DDIM);
        acc.x += al * ov.x; acc.y += al * ov.y;
        acc.z += al * ov.z; acc.w += al * ov.w;
    }
    float* d = Xres + (size_t)r * DDIM + lane * 4;
    atomicAdd(d + 0, scale * acc.x);
    atomicAdd(d + 1, scale * acc.y);
    atomicAdd(d + 2, scale * acc.z);
    atomicAdd(d + 3, scale * acc.w);
}

__global__ void bias_kernel(float* Xres, const float* __restrict__ bias, int total) {
    int i = blockIdx.x * blockDim.x + threadIdx.x;
    if (i < total) Xres[i] += bias[i & (DDIM - 1)];
}

// ---------- launch ----------
static inline int cdiv(long long a, long long b) { return (int)((a + b - 1) / b); }

extern "C" void kernel_launch(void* const* d_in, const int* in_sizes, int n_in,
                              void* d_out, int out_size, void* d_ws, size_t ws_size,
                              hipStream_t stream) {
    (void)n_in; (void)out_size; (void)ws_size;
    const float* X    = (const float*)d_in[0];
    const float* E    = (const float*)d_in[1];
    const int*   adj  = (const int*)  d_in[2];
    const float* av   = (const float*)d_in[3];
    const float* Wn   = (const float*)d_in[4];
    const float* bn   = (const float*)d_in[5];
    const float* We   = (const float*)d_in[6];
    const float* be   = (const float*)d_in[7];
    const float* Wlin = (const float*)d_in[8];
    const float* att  = (const float*)d_in[9];
    const float* bcv  = (const float*)d_in[10];

    const int N   = in_sizes[0] / DDIM;
    const int EN  = in_sizes[1] / DDIM;
    const int NNZ = in_sizes[3];
    const int* rowp = adj;
    const int* colp = adj + NNZ;

    float* Xres = (float*)d_out;                    // [N,128]
    float* Eres = (float*)d_out + (size_t)N * DDIM; // [EN,128]

    // workspace carve (all chunks multiples of 16B)
    float* wsf = (float*)d_ws;
    size_t off = 0;
    float* Xt    = wsf + off; off += (size_t)N  * DDIM;
    float* Etmp  = wsf + off; off += (size_t)EN * DDIM;
    float* agg   = wsf + off; off += (size_t)EN * DDIM;
    float* edeg  = wsf + off; off += (size_t)EN;
    float* DnV   = wsf + off; off += (size_t)N;
    float* BnV   = wsf + off; off += (size_t)EN;
    float* x     = wsf + off; off += (size_t)N  * HF;
    float* e     = wsf + off; off += (size_t)EN * HF;
    float* xa    = wsf + off; off += (size_t)N  * HHEADS;
    float* ea    = wsf + off; off += (size_t)EN * HHEADS;
    float* alpha = wsf + off; off += (size_t)NNZ * HHEADS;
    int*   menc  = (int*)(wsf + off); off += (size_t)N * HHEADS;
    float* ssum  = wsf + off; off += (size_t)N * HHEADS;
    float* oute  = wsf + off; off += (size_t)EN * HF;

    // zero-init accumulators (capturable)
    hipMemsetAsync(agg,  0, (size_t)EN * DDIM * 4, stream);
    hipMemsetAsync(edeg, 0, (size_t)EN * 4, stream);
    hipMemsetAsync(DnV,  0, (size_t)N  * 4, stream);
    hipMemsetAsync(BnV,  0, (size_t)EN * 4, stream);
    hipMemsetAsync(ssum, 0, (size_t)N * HHEADS * 4, stream);
    hipMemsetAsync(oute, 0, (size_t)EN * HF * 4, stream);
    hipMemsetAsync(Xres, 0, (size_t)N * DDIM * 4, stream);
    fill_int_kernel<<<cdiv((size_t)N * HHEADS, 256), 256, 0, stream>>>(menc, INT_MIN, N * HHEADS);

    // degrees
    deg_kernel<<<cdiv(NNZ, 256), 256, 0, stream>>>(rowp, colp, av, edeg, BnV, DnV, NNZ);

    // X_trans = X@Wn + bn
    {
        long long tiles = (long long)((N + 15) / 16) * (DDIM / 16);
        wmma_gemm_kernel<<<cdiv(tiles, 4), 128, 0, stream>>>(X, Wn, bn, Xt, N, DDIM);
    }
    // agg scatter
    scatter_agg_kernel<<<cdiv(NNZ, 8), 256, 0, stream>>>(rowp, colp, av, Xt, agg, NNZ);
    // Etmp = E@We + be
    {
        long long tiles = (long long)((EN + 15) / 16) * (DDIM / 16);
        wmma_gemm_kernel<<<cdiv(tiles, 4), 128, 0, stream>>>(E, We, be, Etmp, EN, DDIM);
    }
    // E_res
    eres_kernel<<<cdiv((size_t)EN * DDIM, 256), 256, 0, stream>>>(agg, edeg, Etmp, E, Eres, EN);

    // x = X@Wlin ; e = E_res@Wlin
    {
        long long tiles = (long long)((N + 15) / 16) * (HF / 16);
        wmma_gemm_kernel<<<cdiv(tiles, 4), 128, 0, stream>>>(X, Wlin, nullptr, x, N, HF);
    }
    {
        long long tiles = (long long)((EN + 15) / 16) * (HF / 16);
        wmma_gemm_kernel<<<cdiv(tiles, 4), 128, 0, stream>>>(Eres, Wlin, nullptr, e, EN, HF);
    }

    // attention scalars
    attdot_kernel<<<cdiv((size_t)N * HHEADS, 256), 256, 0, stream>>>(x, att, 0, xa, N);
    attdot_kernel<<<cdiv((size_t)EN * HHEADS, 256), 256, 0, stream>>>(e, att, DDIM, ea, EN);

    // segment softmax over row
    alpha_max_kernel<<<cdiv((size_t)NNZ * HHEADS, 256), 256, 0, stream>>>(rowp, colp, xa, ea, alpha, menc, NNZ);
    alpha_exp_kernel<<<cdiv((size_t)NNZ * HHEADS, 256), 256, 0, stream>>>(rowp, alpha, menc, ssum, NNZ);
    alpha_norm_kernel<<<cdiv((size_t)NNZ * HHEADS, 256), 256, 0, stream>>>(rowp, alpha, ssum, NNZ);

    // degree inverses
    inv_kernel<<<cdiv(N, 256), 256, 0, stream>>>(DnV, N);
    inv_kernel<<<cdiv(EN, 256), 256, 0, stream>>>(BnV, EN);

    // message passes
    m1_kernel<<<cdiv(NNZ, 8), 256, 0, stream>>>(rowp, colp, alpha, x, oute, NNZ);
    m2_kernel<<<cdiv(NNZ, 8), 256, 0, stream>>>(rowp, colp, alpha, oute, DnV, BnV, Xres, NNZ);

    // + bias_conv
    bias_kernel<<<cdiv((size_t)N * DDIM, 256), 256, 0, stream>>>(Xres, bcv, N * DDIM);
}